// APPNP_Net_78030965834312
// MI455X (gfx1250) — compile-verified
//
#include <hip/hip_runtime.h>
#include <math.h>

// Problem constants (match reference)
#define N_NODES 100000
#define F_IN    128
#define HIDDEN  256
#define CLASSES 47
#define CPAD    48          // padded class stride; column 47 is kept == 0.0f
#define EDGES   1600000
#define KITER   10
#define ALPHA   0.1f
#define NC      (N_NODES * CPAD)

typedef float v2f __attribute__((ext_vector_type(2)));
typedef float v8f __attribute__((ext_vector_type(8)));

// ---------------------------------------------------------------------------
// Fused MLP: h = relu(x @ W1 + b1) @ W2 + b2   -> h[N][CPAD] (col 47 = 0)
// Block = 128 threads = 4 waves; each wave computes a 16-row tile using
// V_WMMA_F32_16X16X4_F32 (fp32 A/B, fp32 accum) per CDNA5 ISA layouts:
//   A (16x4):  lane L holds M = L&15, K = 2*(L>>4) + {0,1}   (2 VGPRs)
//   B (4x16):  lane L holds N = L&15, K = 2*(L>>4) + {0,1}   (2 VGPRs)
//   C/D 16x16: VGPR i, lanes 0-15 -> M=i, lanes 16-31 -> M=i+8, N = L&15
// ---------------------------------------------------------------------------
__global__ __launch_bounds__(128) void mlp_kernel(
    const float* __restrict__ x,
    const float* __restrict__ W1, const float* __restrict__ b1,
    const float* __restrict__ W2, const float* __restrict__ b2,
    float* __restrict__ h)
{
    __shared__ float lds[4][16][HIDDEN];   // 64 KB; per-wave 16x256 relu tile

    const int wave = threadIdx.x >> 5;
    const int lane = threadIdx.x & 31;
    const int half = lane >> 4;
    const int col  = lane & 15;
    const int m    = col;
    const int rowBase = blockIdx.x * 64 + wave * 16;

    int arow = rowBase + m;
    if (arow >= N_NODES) arow = N_NODES - 1;   // clamp loads; stores guarded
    const float* xrow = x + (size_t)arow * F_IN;

    // ---------------- GEMM1: [16 x 128] @ [128 x 256] ----------------
    for (int ct = 0; ct < 16; ++ct) {
        const int n = ct * 16 + col;
        v8f acc = {};
        #pragma unroll
        for (int k0 = 0; k0 < F_IN; k0 += 4) {
            const int k = k0 + 2 * half;
            v2f a = *(const v2f*)(xrow + k);
            v2f b;
            b.x = W1[(size_t)(k)     * HIDDEN + n];
            b.y = W1[(size_t)(k + 1) * HIDDEN + n];
            acc = __builtin_amdgcn_wmma_f32_16x16x4_f32(
                      false, a, false, b, (short)0, acc, false, false);
        }
        const float bias = b1[n];
        #pragma unroll
        for (int i = 0; i < 8; ++i) {
            float v = acc[i] + bias;
            lds[wave][i + 8 * half][n] = v > 0.0f ? v : 0.0f;   // ReLU
        }
    }

    // ---------------- GEMM2: [16 x 256] @ [256 x 47(+1 pad)] ----------------
    for (int ct2 = 0; ct2 < 3; ++ct2) {
        const int n2 = ct2 * 16 + col;
        const bool valid = (n2 < CLASSES);
        v8f acc = {};
        #pragma unroll
        for (int k0 = 0; k0 < HIDDEN; k0 += 4) {
            const int k = k0 + 2 * half;
            v2f a = *(const v2f*)&lds[wave][m][k];
            v2f b;
            b.x = valid ? W2[(size_t)(k)     * CLASSES + n2] : 0.0f;
            b.y = valid ? W2[(size_t)(k + 1) * CLASSES + n2] : 0.0f;
            acc = __builtin_amdgcn_wmma_f32_16x16x4_f32(
                      false, a, false, b, (short)0, acc, false, false);
        }
        const float bias = valid ? b2[n2] : 0.0f;
        #pragma unroll
        for (int i = 0; i < 8; ++i) {
            const int r = rowBase + i + 8 * half;
            if (r < N_NODES)                    // pad column stores 0.0f
                h[(size_t)r * CPAD + n2] = acc[i] + bias;
        }
    }
}

// ---------------------------------------------------------------------------
// CSR construction (once per launch): cnt -> scan -> fill
// ---------------------------------------------------------------------------
__global__ __launch_bounds__(256) void zero_int_kernel(int* __restrict__ p, int n) {
    int i = blockIdx.x * 256 + threadIdx.x;
    if (i < n) p[i] = 0;
}

__global__ __launch_bounds__(256) void count_kernel(
    const int* __restrict__ dstIdx, int* __restrict__ cnt) {
    int e = blockIdx.x * 256 + threadIdx.x;
    if (e < EDGES) atomicAdd(&cnt[dstIdx[e]], 1);
}

// Single-workgroup exclusive prefix scan: rowstart[0..N] from cnt[0..N-1].
// 1024 threads x 98-element chunks (1024*98 = 100352 >= 100001).
__global__ __launch_bounds__(1024) void scan_kernel(
    const int* __restrict__ cnt, int* __restrict__ rowstart) {
    __shared__ int partial[1024];
    const int t  = threadIdx.x;
    const int CH = 98;
    const int base = t * CH;

    int s = 0;
    for (int i = 0; i < CH; ++i) {
        int idx = base + i;
        if (idx < N_NODES) s += cnt[idx];
    }
    partial[t] = s;
    __syncthreads();
    // Hillis-Steele inclusive scan over 1024 partials
    #pragma unroll
    for (int off = 1; off < 1024; off <<= 1) {
        int v = (t >= off) ? partial[t - off] : 0;
        __syncthreads();
        partial[t] += v;
        __syncthreads();
    }
    int run = (t == 0) ? 0 : partial[t - 1];   // exclusive chunk offset
    for (int i = 0; i < CH; ++i) {
        int idx = base + i;
        if (idx <= N_NODES) {
            rowstart[idx] = run;
            if (idx < N_NODES) run += cnt[idx];
        }
    }
}

__global__ __launch_bounds__(256) void copy_int_kernel(
    const int* __restrict__ a, int* __restrict__ b, int n) {
    int i = blockIdx.x * 256 + threadIdx.x;
    if (i < n) b[i] = a[i];
}

__global__ __launch_bounds__(256) void fill_kernel(
    const int* __restrict__ srcIdx, const int* __restrict__ dstIdx,
    int* __restrict__ pos, int* __restrict__ csr_src) {
    int e = blockIdx.x * 256 + threadIdx.x;
    if (e >= EDGES) return;
    int d = dstIdx[e];
    int idx = atomicAdd(&pos[d], 1);
    csr_src[idx] = srcIdx[e];
}

__global__ __launch_bounds__(256) void dinv_kernel(
    const int* __restrict__ cnt, float* __restrict__ dinv) {
    int i = blockIdx.x * 256 + threadIdx.x;
    if (i < N_NODES) dinv[i] = rsqrtf((float)cnt[i] + 1.0f);  // +1 self loop
}

// zs0 = dinv o h  (z0 = h)
__global__ __launch_bounds__(256) void zsinit_kernel(
    const float* __restrict__ dinv, const float* __restrict__ hbuf,
    float* __restrict__ zs) {
    int i = blockIdx.x * 256 + threadIdx.x;
    if (i < NC) zs[i] = dinv[i / CPAD] * hbuf[i];
}

// ---------------------------------------------------------------------------
// One APPNP round, atomic-free gather over CSR in-edges.
//   z_{k+1}[n][c] = 0.9 * dinv[n] * ( sum_{s in N(n)} zs[s][c] + zs[n][c] )
//                 + 0.1 * h[n][c]
//   zs_{k+1}      = dinv[n] * z_{k+1}
// zs (19 MB) and csr (6.4 MB) stay resident in the 192 MB L2.
// ---------------------------------------------------------------------------
__global__ __launch_bounds__(256) void gather_kernel(
    const float* __restrict__ zs_in, const float* __restrict__ hbuf,
    const float* __restrict__ dinv, const int* __restrict__ rowstart,
    const int* __restrict__ csr_src,
    float* __restrict__ zs_out, float* __restrict__ z_out) {
    int i = blockIdx.x * 256 + threadIdx.x;
    if (i >= NC) return;
    int n = i / CPAD;
    int c = i - n * CPAD;
    int jb = rowstart[n];
    int je = rowstart[n + 1];
    float acc = 0.0f;
    for (int j = jb; j < je; ++j) {
        int s = csr_src[j];
        acc += zs_in[(size_t)s * CPAD + c];
    }
    float dn = dinv[n];
    float znew = (1.0f - ALPHA) * dn * (acc + zs_in[i]) + ALPHA * hbuf[i];
    z_out[i]  = znew;
    zs_out[i] = dn * znew;
}

// Wave-per-row log_softmax over 47 classes (wave32 shuffles)
__global__ __launch_bounds__(256) void logsoftmax_kernel(
    const float* __restrict__ z, float* __restrict__ out) {
    const int wave = threadIdx.x >> 5;
    const int lane = threadIdx.x & 31;
    const int row  = blockIdx.x * 8 + wave;
    if (row >= N_NODES) return;
    const float* zr = z + (size_t)row * CPAD;
    const int c2 = lane + 32;
    float v0 = zr[lane];
    float v1 = (c2 < CLASSES) ? zr[c2] : -INFINITY;
    float mx = fmaxf(v0, v1);
    #pragma unroll
    for (int off = 16; off > 0; off >>= 1)
        mx = fmaxf(mx, __shfl_xor(mx, off, 32));
    float s = __expf(v0 - mx) + ((c2 < CLASSES) ? __expf(v1 - mx) : 0.0f);
    #pragma unroll
    for (int off = 16; off > 0; off >>= 1)
        s += __shfl_xor(s, off, 32);
    float lse = __logf(s);
    float* orow = out + (size_t)row * CLASSES;
    orow[lane] = v0 - mx - lse;
    if (c2 < CLASSES) orow[c2] = v1 - mx - lse;
}

// ---------------------------------------------------------------------------
extern "C" void kernel_launch(void* const* d_in, const int* in_sizes, int n_in,
                              void* d_out, int out_size, void* d_ws, size_t ws_size,
                              hipStream_t stream) {
    const float* x  = (const float*)d_in[0];
    const int*   ei = (const int*)d_in[1];        // (2, E) flat: [src | dst]
    const float* W1 = (const float*)d_in[2];
    const float* b1 = (const float*)d_in[3];
    const float* W2 = (const float*)d_in[4];
    const float* b2 = (const float*)d_in[5];
    float* out = (float*)d_out;

    const int* srcIdx = ei;
    const int* dstIdx = ei + EDGES;

    // Workspace layout: h | z | zs0 | zs1 (floats) | dinv | cnt | rowstart | pos | csr
    float* h    = (float*)d_ws;
    float* z    = h   + (size_t)NC;
    float* zs0  = z   + (size_t)NC;
    float* zs1  = zs0 + (size_t)NC;
    float* dinv = zs1 + (size_t)NC;
    int* cnt      = (int*)(dinv + N_NODES);
    int* rowstart = cnt + N_NODES;            // N+1 entries
    int* pos      = rowstart + (N_NODES + 1);
    int* csr_src  = pos + N_NODES;            // E entries

    // 1) Fused MLP with fp32 WMMA
    mlp_kernel<<<(N_NODES + 63) / 64, 128, 0, stream>>>(x, W1, b1, W2, b2, h);

    // 2) CSR build: counts -> exclusive scan -> positional fill
    zero_int_kernel<<<(N_NODES + 255) / 256, 256, 0, stream>>>(cnt, N_NODES);
    count_kernel<<<(EDGES + 255) / 256, 256, 0, stream>>>(dstIdx, cnt);
    scan_kernel<<<1, 1024, 0, stream>>>(cnt, rowstart);
    copy_int_kernel<<<(N_NODES + 255) / 256, 256, 0, stream>>>(rowstart, pos, N_NODES);
    fill_kernel<<<(EDGES + 255) / 256, 256, 0, stream>>>(srcIdx, dstIdx, pos, csr_src);

    // 3) Normalization: dinv = (deg+1)^-1/2 ; zs0 = dinv o h
    dinv_kernel<<<(N_NODES + 255) / 256, 256, 0, stream>>>(cnt, dinv);
    zsinit_kernel<<<(NC + 255) / 256, 256, 0, stream>>>(dinv, h, zs0);

    // 4) K atomic-free gather rounds (ping-pong zs buffers)
    float* zin  = zs0;
    float* zout = zs1;
    for (int k = 0; k < KITER; ++k) {
        gather_kernel<<<(NC + 255) / 256, 256, 0, stream>>>(
            zin, h, dinv, rowstart, csr_src, zout, z);
        float* t = zin; zin = zout; zout = t;
    }

    // 5) log_softmax -> d_out [N, 47]
    logsoftmax_kernel<<<(N_NODES + 7) / 8, 256, 0, stream>>>(z, out);
}